// MPSLayer_7816840479009
// MI455X (gfx1250) — compile-verified
//
#include <hip/hip_runtime.h>
#include <cstddef>

// MPS layer, collapsed algebraically (exact up to fp reassociation):
//   out[b,o] = s_b * v[o] + bias[o]
//   s_b  = prod_i x[b,i]
//   m    = ones_row @ C_0 @ ... @ C_255   (tree-reassociated, WMMA f32 16x16x4)
//   v[o] = sum_r m[r] * sum_l proj[r,l,o]

typedef float v2f __attribute__((ext_vector_type(2)));
typedef float v8f __attribute__((ext_vector_type(8)));

#define BATCH 1024
#define IN_N  256
#define DD    32
#define OUT_O 512

// ---------------------------------------------------------------------------
// 32x32 * 32x32 f32 matmul on one wave32 via V_WMMA_F32_16X16X4_F32.
// A,B,D are row-major 32x32 float arrays (LDS or global). D==A is allowed:
// all 4 output tiles are fully accumulated in VGPRs before any store.
// Operand layouts (ISA 7.12.2):
//   A 16x4 : lanes 0-15 -> M=lane, vgpr0=K0 vgpr1=K1 ; lanes 16-31 -> K2,K3
//   B 4x16 : mirrored   ; C/D 16x16: vgpr j -> (M=j | j+8, N=lane&15)
// ---------------------------------------------------------------------------
__device__ __forceinline__ void mm32(const float* A, const float* B, float* D) {
  const int lane = threadIdx.x & 31;
  const int row  = lane & 15;
  const int h    = lane >> 4;   // 0: lanes 0-15 (K 0,1)   1: lanes 16-31 (K 2,3)
  v8f acc[2][2];
#pragma unroll
  for (int ti = 0; ti < 2; ++ti) {
#pragma unroll
    for (int tj = 0; tj < 2; ++tj) {
      v8f c = {0.f, 0.f, 0.f, 0.f, 0.f, 0.f, 0.f, 0.f};
#pragma unroll
      for (int kb = 0; kb < 8; ++kb) {
        v2f a, b;
        a.x = A[(ti * 16 + row) * 32 + kb * 4 + 2 * h + 0];
        a.y = A[(ti * 16 + row) * 32 + kb * 4 + 2 * h + 1];
        b.x = B[(kb * 4 + 2 * h + 0) * 32 + tj * 16 + row];
        b.y = B[(kb * 4 + 2 * h + 1) * 32 + tj * 16 + row];
        c = __builtin_amdgcn_wmma_f32_16x16x4_f32(false, a, false, b,
                                                  (short)0, c, false, false);
      }
      acc[ti][tj] = c;
    }
  }
#pragma unroll
  for (int ti = 0; ti < 2; ++ti)
#pragma unroll
    for (int tj = 0; tj < 2; ++tj)
#pragma unroll
      for (int j = 0; j < 8; ++j)
        D[(ti * 16 + j + 8 * h) * 32 + tj * 16 + row] = acc[ti][tj][j];
}

// K1: s[b] = prod_i x[b,i].  One wave per batch row, shfl butterfly product.
__global__ void __launch_bounds__(256) k_batch_prod(const float* __restrict__ x,
                                                    float* __restrict__ s) {
  const int wave = threadIdx.x >> 5;
  const int lane = threadIdx.x & 31;
  const int b    = blockIdx.x * 8 + wave;
  const float* row = x + (size_t)b * IN_N;
  float p = 1.0f;
#pragma unroll
  for (int j = 0; j < IN_N / 32; ++j) p *= row[j * 32 + lane];
#pragma unroll
  for (int off = 16; off > 0; off >>= 1) p *= __shfl_xor(p, off, 32);
  if (lane == 0) s[b] = p;
}

// K2: m = ones_row @ C_0 @ ... @ C_255, as a parallel tree of WMMA matmuls.
// 16 waves (512 thr): wave w chains 16 cores sequentially in its LDS slot,
// then 4 tree levels pairwise-multiply the 16 partial products, then wave 0
// column-sums the final product (ones_row @ P).
__global__ void __launch_bounds__(512) k_chain(const float* __restrict__ cores,
                                               float* __restrict__ m_out) {
  __shared__ float lds[16 * 1024];          // 16 slots of 32x32 f32 (64 KB)
  const int w    = threadIdx.x >> 5;
  const int lane = threadIdx.x & 31;
  float* slot = lds + w * 1024;

  // init slot with first core of this wave's chunk (coalesced copy)
  const float* g0 = cores + (size_t)(w * 16) * 1024;
#pragma unroll
  for (int i = 0; i < 32; ++i) slot[i * 32 + lane] = g0[i * 32 + lane];

  // 15 in-order multiplies: slot = slot @ C_{16w+t}
  for (int t = 1; t < 16; ++t)
    mm32(slot, cores + (size_t)(w * 16 + t) * 1024, slot);
  __syncthreads();

  // tree: 16 -> 8 -> 4 -> 2 -> 1 (order-preserving pairing, left slot = dst)
  for (int s = 1; s <= 8; s <<= 1) {
    const int nw = 8 / s;
    if (w < nw) {
      float* L = lds + (size_t)(2 * w * s) * 1024;
      float* R = L + (size_t)s * 1024;
      mm32(L, R, L);
    }
    __syncthreads();
  }

  // m[c] = sum_r P[r][c]   (ones_row @ P)
  if (threadIdx.x < 32) {
    float acc = 0.f;
#pragma unroll
    for (int r = 0; r < 32; ++r) acc += lds[r * 32 + lane];
    m_out[lane] = acc;
  }
}

// K3a: partial[r][o] = m[r] * sum_l proj[r,l,o].  One block per r (coalesced
// over o, spreads the 2 MB projection read across 32 WGPs).
__global__ void __launch_bounds__(256) k_proj_partial(const float* __restrict__ proj,
                                                      const float* __restrict__ m,
                                                      float* __restrict__ part) {
  const int r = blockIdx.x;
  const int o = threadIdx.x;            // handles o and o+256
  const float mr = m[r];
  const float* p = proj + (size_t)r * DD * OUT_O;
  float a0 = 0.f, a1 = 0.f;
#pragma unroll 8
  for (int l = 0; l < DD; ++l) {
    a0 += p[l * OUT_O + o];
    a1 += p[l * OUT_O + o + 256];
  }
  part[r * OUT_O + o]       = mr * a0;
  part[r * OUT_O + o + 256] = mr * a1;
}

// K3b: v[o] = sum_r partial[r][o]
__global__ void __launch_bounds__(512) k_proj_reduce(const float* __restrict__ part,
                                                     float* __restrict__ v) {
  const int o = threadIdx.x;
  float acc = 0.f;
#pragma unroll
  for (int r = 0; r < DD; ++r) acc += part[r * OUT_O + o];
  v[o] = acc;
}

// K4: out[b,o] = s[b]*v[o] + bias[o].  One 16x16 tile per wave via a single
// WMMA (A col K=0 = s chunk, B row K=0 = v chunk, C = bias) -> exact f32
// rank-1 update; stores are the dominant (coalesced) 2 MB stream.
__global__ void __launch_bounds__(256) k_outer_wmma(const float* __restrict__ s,
                                                    const float* __restrict__ v,
                                                    const float* __restrict__ bias,
                                                    float* __restrict__ out) {
  const int wave = threadIdx.x >> 5;
  const int lane = threadIdx.x & 31;
  const int tile = blockIdx.x * 8 + wave;   // 2048 tiles total
  const int bt   = tile >> 5;               // 0..63  (batch tile)
  const int ot   = tile & 31;               // 0..31  (output tile)
  const int row  = lane & 15;
  const int h    = lane >> 4;
  const bool lo  = (h == 0);

  const float sv = s[bt * 16 + row];        // in-bounds for all lanes
  const float vv = v[ot * 16 + row];
  const float bb = bias[ot * 16 + row];

  v2f A; A.x = lo ? sv : 0.0f; A.y = 0.0f;  // only K=0 column populated
  v2f B; B.x = lo ? vv : 0.0f; B.y = 0.0f;  // only K=0 row populated
  v8f C = {bb, bb, bb, bb, bb, bb, bb, bb};
  v8f D = __builtin_amdgcn_wmma_f32_16x16x4_f32(false, A, false, B,
                                                (short)0, C, false, false);
  float* dst = out + (size_t)(bt * 16) * OUT_O + ot * 16 + row;
#pragma unroll
  for (int j = 0; j < 8; ++j)
    dst[(size_t)(j + 8 * h) * OUT_O] = D[j];
}

extern "C" void kernel_launch(void* const* d_in, const int* in_sizes, int n_in,
                              void* d_out, int out_size, void* d_ws, size_t ws_size,
                              hipStream_t stream) {
  (void)in_sizes; (void)n_in; (void)out_size; (void)ws_size;
  const float* x     = (const float*)d_in[0];   // [1024,256]
  const float* cores = (const float*)d_in[1];   // [256,32,32]
  const float* proj  = (const float*)d_in[2];   // [32,32,512]
  const float* bias  = (const float*)d_in[3];   // [512]
  float* out = (float*)d_out;                   // [1024,512]

  float* ws   = (float*)d_ws;
  float* ws_s = ws;                  // 1024 floats
  float* ws_m = ws_s + BATCH;        // 32 floats
  float* ws_v = ws_m + DD;           // 512 floats
  float* ws_p = ws_v + OUT_O;        // 32*512 floats (~64 KB); ~72 KB total

  k_batch_prod <<<BATCH / 8, 256, 0, stream>>>(x, ws_s);
  k_chain      <<<1, 512, 0, stream>>>(cores, ws_m);
  k_proj_partial<<<DD, 256, 0, stream>>>(proj, ws_m, ws_p);
  k_proj_reduce<<<1, OUT_O, 0, stream>>>(ws_p, ws_v);
  k_outer_wmma <<<256, 256, 0, stream>>>(ws_s, ws_v, bias, out);
}